// emssemble_model_45861660786781
// MI455X (gfx1250) — compile-verified
//
#include <hip/hip_runtime.h>
#include <math.h>

#define BATCH 256
#define NNODE 256
#define FIN   600
#define HID   128
#define GED   128
#define DEMO  6
#define NCLS  2
#define PE    500
#define GE    5000
#define GFEAT (GED + DEMO)   // 134

typedef float v2f __attribute__((ext_vector_type(2)));
typedef float v8f __attribute__((ext_vector_type(8)));

#define KC    64     // K-chunk staged in LDS
#define LDSP  144    // pair-row stride (floats2) -> 2*144 dwords = 288 ≡ 32 (mod 64) banks

// ---------------------------------------------------------------------------
// WMMA GEMM with N fixed at 128:  C[M,128] = A[M,K] @ Bw[K,128]
// Requirements: M % 16 == 0, K % 4 == 0.
// Block = 128 threads (4 waves); each wave computes a 16-row x 128-col stripe
// with v_wmma_f32_16x16x4_f32.  Weights are staged per-block into LDS as
// (k,k+1) row-pairs so each B fragment is a single ds_load_b64; the A fragment
// is a single global b64 per K-step, reused across the 8 N-tiles.
//
// VGPR layouts per CDNA5 ISA 7.12.2:
//   A (16x4 f32):  lanes 0-15 hold K={0,1}, lanes 16-31 hold K={2,3}; M = lane&15
//   B (4x16 f32):  lanes 0-15 hold rows K={0,1}, lanes 16-31 rows K={2,3}; N = lane&15
//   C (16x16 f32): VGPR r -> row r (lanes 0-15) / row r+8 (lanes 16-31); N = lane&15
// ---------------------------------------------------------------------------
__global__ void __launch_bounds__(128)
wmma_gemm_n128(const float* __restrict__ A, const float* __restrict__ Bw,
               float* __restrict__ C, int M, int K) {
  __shared__ v2f ldsB[(KC / 2) * LDSP];   // 32*144*8 = 36,864 bytes

  const int tid  = threadIdx.x;
  const int lane = tid & 31;
  const int wave = (blockIdx.x * blockDim.x + tid) >> 5;
  const int mBase = wave * 16;
  const int mUse  = (mBase < M) ? mBase : 0;   // keep whole block in syncs

  const int half = lane >> 4;       // 0 => lanes 0-15, 1 => lanes 16-31
  const int lid  = lane & 15;
  const int kk   = half * 2;

  v8f acc[8] = {};
  const float* arow = A + (size_t)(mUse + lid) * K;

  for (int kc = 0; kc < K; kc += KC) {
    const int kcLen = (K - kc < KC) ? (K - kc) : KC;   // multiple of 4

    // hint the next A chunk into cache while we stage B
    __builtin_prefetch(arow + kc + KC, 0, 0);

    // cooperative stage of B[kc .. kc+kcLen) into LDS as (k,k+1) column pairs
    for (int idx = tid; idx < (kcLen / 2) * 128; idx += 128) {
      const int kp = idx >> 7;      // pair index within chunk
      const int n  = idx & 127;
      v2f p;
      p.x = Bw[(size_t)(kc + 2 * kp + 0) * 128 + n];
      p.y = Bw[(size_t)(kc + 2 * kp + 1) * 128 + n];
      ldsB[kp * LDSP + n] = p;
    }
    __syncthreads();

    for (int k = 0; k < kcLen; k += 4) {
      const v2f a = *(const v2f*)(arow + kc + k + kk);     // global b64
      const v2f* brow = ldsB + ((k >> 1) + half) * LDSP + lid;
#pragma unroll
      for (int t = 0; t < 8; ++t) {
        const v2f b = brow[t * 16];                        // ds_load_b64
        acc[t] = __builtin_amdgcn_wmma_f32_16x16x4_f32(
            false, a, false, b, (short)0, acc[t], false, false);
      }
    }
    __syncthreads();
  }

  if (mBase < M) {
    const int rowBase = mBase + half * 8;
#pragma unroll
    for (int t = 0; t < 8; ++t) {
      float* cp = C + (size_t)rowBase * 128 + t * 16 + lid;
#pragma unroll
      for (int r = 0; r < 8; ++r) cp[(size_t)r * 128] = acc[t][r];
    }
  }
}

// ---------------------------------------------------------------------------
// Degree -> deg^{-1/2} per graph.  Self-loop folded in via cnt init = 1.
// blockIdx.x = graph id, blockDim.x == NNODE (256).
// ---------------------------------------------------------------------------
__global__ void deg_kernel(const int* __restrict__ dstBase, int graphStride,
                           int numEdges, float* __restrict__ dinv) {
  __shared__ int cnt[NNODE];
  const int g = blockIdx.x, t = threadIdx.x;
  cnt[t] = 1;
  __syncthreads();
  const int* d = dstBase + (size_t)g * graphStride;
  for (int e = t; e < numEdges; e += blockDim.x) atomicAdd(&cnt[d[e]], 1);
  __syncthreads();
  dinv[g * NNODE + t] = rsqrtf((float)cnt[t]);
}

// ---------------------------------------------------------------------------
// Patient-level GCN aggregation: out[b,n,:] = sum_{e:s->n} dinv[s]dinv[n] h[b,s,:]
//                                            + dinv[n]^2 h[b,n,:] + bias ; opt ReLU
// One block per patient, thread f owns feature column f end-to-end -> no atomics.
// ---------------------------------------------------------------------------
__global__ void __launch_bounds__(HID)
patient_aggregate(const float* __restrict__ h, const int* __restrict__ pei,
                  const float* __restrict__ dinv, const float* __restrict__ bias,
                  float* __restrict__ out, int relu) {
  const int b = blockIdx.x;
  const int f = threadIdx.x;
  const float* hb = h   + (size_t)b * NNODE * HID;
  float*       ob = out + (size_t)b * NNODE * HID;
  const float* dv = dinv + b * NNODE;
  const int* srcs = pei + (size_t)b * 2 * PE;
  const int* dsts = srcs + PE;

  for (int n = 0; n < NNODE; ++n) ob[n * HID + f] = 0.f;
  for (int e = 0; e < PE; ++e) {
    const int s = srcs[e], d = dsts[e];
    ob[d * HID + f] += dv[s] * dv[d] * hb[s * HID + f];
  }
  const float bf = bias[f];
  for (int n = 0; n < NNODE; ++n) {
    float v = ob[n * HID + f] + dv[n] * dv[n] * hb[n * HID + f] + bf;
    ob[n * HID + f] = relu ? fmaxf(v, 0.f) : v;
  }
}

// Group-level aggregation over the B=256 patient graph. Single block,
// thread f owns column f; F in {64,32,16,2}.
__global__ void group_aggregate(const float* __restrict__ h,
                                const int* __restrict__ gsrc,
                                const int* __restrict__ gdst,
                                const float* __restrict__ dinv,
                                const float* __restrict__ bias,
                                float* __restrict__ out, int F, int relu) {
  const int f = threadIdx.x;
  if (f >= F) return;
  for (int n = 0; n < BATCH; ++n) out[n * F + f] = 0.f;
  for (int e = 0; e < GE; ++e) {
    const int s = gsrc[e], d = gdst[e];
    out[d * F + f] += dinv[s] * dinv[d] * h[s * F + f];
  }
  const float bf = bias[f];
  for (int n = 0; n < BATCH; ++n) {
    float v = out[n * F + f] + dinv[n] * dinv[n] * h[n * F + f] + bf;
    out[n * F + f] = relu ? fmaxf(v, 0.f) : v;
  }
}

// Max-pool over nodes: pooled[b,f] = max_n h[b,n,f]
__global__ void __launch_bounds__(HID)
maxpool_kernel(const float* __restrict__ h, float* __restrict__ pooled) {
  const int b = blockIdx.x, f = threadIdx.x;
  const float* hb = h + (size_t)b * NNODE * HID;
  float m = hb[f];
  for (int n = 1; n < NNODE; ++n) m = fmaxf(m, hb[n * HID + f]);
  pooled[b * HID + f] = m;
}

// feat[b] = concat(lin[b] + plb, demographic[b])
__global__ void build_feat(const float* __restrict__ lin, const float* __restrict__ plb,
                           const float* __restrict__ demo, float* __restrict__ feat) {
  const int b = blockIdx.x, t = threadIdx.x;
  if (t < GED)        feat[b * GFEAT + t] = lin[b * GED + t] + plb[t];
  else if (t < GFEAT) feat[b * GFEAT + t] = demo[b * DEMO + (t - GED)];
}

// Tiny scalar GEMM for the group net: C[M,Nn] = A[M,K] @ W[K,Nn]
__global__ void small_gemm(const float* __restrict__ A, const float* __restrict__ W,
                           float* __restrict__ C, int M, int K, int Nn) {
  const int idx = blockIdx.x * blockDim.x + threadIdx.x;
  if (idx >= M * Nn) return;
  const int m = idx / Nn, n = idx % Nn;
  float s = 0.f;
  for (int k = 0; k < K; ++k) s += A[m * K + k] * W[k * Nn + n];
  C[idx] = s;
}

// out[0:512)   = log_softmax(gh4)          flattened [256,2]
// out[512:1024)= softmax(feat @ clW + clb) flattened [256,2]
__global__ void final_kernel(const float* __restrict__ gh4, const float* __restrict__ feat,
                             const float* __restrict__ clW, const float* __restrict__ clb,
                             float* __restrict__ out) {
  const int b = blockIdx.x * blockDim.x + threadIdx.x;
  if (b >= BATCH) return;
  const float a0 = gh4[b * 2 + 0], a1 = gh4[b * 2 + 1];
  const float mx = fmaxf(a0, a1);
  const float lse = mx + logf(expf(a0 - mx) + expf(a1 - mx));
  out[b * 2 + 0] = a0 - lse;
  out[b * 2 + 1] = a1 - lse;

  float l0 = clb[0], l1 = clb[1];
  const float* fb = feat + (size_t)b * GFEAT;
  for (int k = 0; k < GFEAT; ++k) {
    const float xv = fb[k];
    l0 += xv * clW[k * 2 + 0];
    l1 += xv * clW[k * 2 + 1];
  }
  const float m2 = fmaxf(l0, l1);
  const float e0 = expf(l0 - m2), e1 = expf(l1 - m2);
  const float s = e0 + e1;
  out[512 + b * 2 + 0] = e0 / s;
  out[512 + b * 2 + 1] = e1 / s;
}

extern "C" void kernel_launch(void* const* d_in, const int* in_sizes, int n_in,
                              void* d_out, int out_size, void* d_ws, size_t ws_size,
                              hipStream_t stream) {
  (void)in_sizes; (void)n_in; (void)out_size; (void)ws_size;
  const float* x    = (const float*)d_in[0];
  const float* demo = (const float*)d_in[1];
  const int*   pei  = (const int*)d_in[2];
  const int*   gei  = (const int*)d_in[3];
  const float* pW1 = (const float*)d_in[4];  const float* pb1 = (const float*)d_in[5];
  const float* pW2 = (const float*)d_in[6];  const float* pb2 = (const float*)d_in[7];
  const float* pW3 = (const float*)d_in[8];  const float* pb3 = (const float*)d_in[9];
  const float* plW = (const float*)d_in[10]; const float* plb = (const float*)d_in[11];
  const float* gW1 = (const float*)d_in[12]; const float* gb1 = (const float*)d_in[13];
  const float* gW2 = (const float*)d_in[14]; const float* gb2 = (const float*)d_in[15];
  const float* gW3 = (const float*)d_in[16]; const float* gb3 = (const float*)d_in[17];
  const float* gW4 = (const float*)d_in[18]; const float* gb4 = (const float*)d_in[19];
  const float* clW = (const float*)d_in[20]; const float* clb = (const float*)d_in[21];
  float* out = (float*)d_out;

  // workspace carve-up (floats)
  float* ws     = (float*)d_ws;
  float* dinvP  = ws;                                        // 256*256
  float* hA     = dinvP + BATCH * NNODE;                     // 256*256*128
  float* hB     = hA + (size_t)BATCH * NNODE * HID;          // 256*256*128
  float* pooled = hB + (size_t)BATCH * NNODE * HID;          // 256*128
  float* feat   = pooled + BATCH * HID;                      // 256*134
  float* gdinv  = feat + BATCH * GFEAT;                      // 256
  float* gt     = gdinv + BATCH;                             // 256*64 max
  float* gh     = gt + BATCH * 64;                           // 256*64 max

  const int MBIG = BATCH * NNODE;   // 65536 rows

  // ---- patient-level GCN ----
  deg_kernel<<<BATCH, NNODE, 0, stream>>>(pei + PE, 2 * PE, PE, dinvP);

  wmma_gemm_n128<<<MBIG / 64, 128, 0, stream>>>(x, pW1, hA, MBIG, FIN);
  patient_aggregate<<<BATCH, HID, 0, stream>>>(hA, pei, dinvP, pb1, hB, 1);

  wmma_gemm_n128<<<MBIG / 64, 128, 0, stream>>>(hB, pW2, hA, MBIG, HID);
  patient_aggregate<<<BATCH, HID, 0, stream>>>(hA, pei, dinvP, pb2, hB, 1);

  wmma_gemm_n128<<<MBIG / 64, 128, 0, stream>>>(hB, pW3, hA, MBIG, HID);
  patient_aggregate<<<BATCH, HID, 0, stream>>>(hA, pei, dinvP, pb3, hB, 1);

  maxpool_kernel<<<BATCH, HID, 0, stream>>>(hB, pooled);
  wmma_gemm_n128<<<BATCH / 64, 128, 0, stream>>>(pooled, plW, hA, BATCH, HID);
  build_feat<<<BATCH, 160, 0, stream>>>(hA, plb, demo, feat);

  // ---- group-level GCN ----
  deg_kernel<<<1, BATCH, 0, stream>>>(gei + GE, 0, GE, gdinv);

  small_gemm<<<(BATCH * 64 + 255) / 256, 256, 0, stream>>>(feat, gW1, gt, BATCH, GFEAT, 64);
  group_aggregate<<<1, 64, 0, stream>>>(gt, gei, gei + GE, gdinv, gb1, gh, 64, 1);

  small_gemm<<<(BATCH * 32 + 255) / 256, 256, 0, stream>>>(gh, gW2, gt, BATCH, 64, 32);
  group_aggregate<<<1, 64, 0, stream>>>(gt, gei, gei + GE, gdinv, gb2, gh, 32, 1);

  small_gemm<<<(BATCH * 16 + 255) / 256, 256, 0, stream>>>(gh, gW3, gt, BATCH, 32, 16);
  group_aggregate<<<1, 64, 0, stream>>>(gt, gei, gei + GE, gdinv, gb3, gh, 16, 1);

  small_gemm<<<(BATCH * 2 + 255) / 256, 256, 0, stream>>>(gh, gW4, gt, BATCH, 16, 2);
  group_aggregate<<<1, 64, 0, stream>>>(gt, gei, gei + GE, gdinv, gb4, gh, 2, 0);

  final_kernel<<<1, BATCH, 0, stream>>>(gh, feat, clW, clb, out);
}